// LIFNodeSTBP_63513976373966
// MI455X (gfx1250) — compile-verified
//
#include <hip/hip_runtime.h>
#include <cstdint>

// LIF forward scan, MI455X (gfx1250).
// x:[B=32, T=16, S=65536] fp32 -> spikes same shape. Pure streaming: 268 MB
// HBM traffic, ~11.5 us roofline at 23.3 TB/s. One thread owns a 4-float
// column and scans T=16 in registers; input is staged through LDS with the
// CDNA5 async data mover (GLOBAL_LOAD_ASYNC_TO_LDS_B128 + ASYNCcnt), spikes
// leave via non-temporal stores.

typedef __attribute__((ext_vector_type(4))) float v4f;  // native vector (not HIP_vector_type)

constexpr int T_STEPS  = 16;
constexpr int S_ELEMS  = 64 * 32 * 32;   // 65536 spatial elements per (b,t)
constexpr int B_BATCH  = 32;
constexpr int NTHREADS = 256;            // 8 waves (wave32)
constexpr int VEC      = 4;              // floats per thread per timestep
constexpr int CHUNK    = NTHREADS * VEC; // 1024 floats of s per block

__global__ __launch_bounds__(NTHREADS)
void lif_scan_async(const float* __restrict__ x, float* __restrict__ out) {
    // One 4KB stage per timestep, 64 KiB total (well under 320 KB/WGP).
    __shared__ v4f smem[T_STEPS][NTHREADS];

    const int tid          = threadIdx.x;
    const int blocks_per_b = S_ELEMS / CHUNK;                 // 64
    const int b            = blockIdx.x / blocks_per_b;
    const int sc           = blockIdx.x - b * blocks_per_b;

    // flat index of (b, t=0, s) for this thread; time stride = S_ELEMS floats
    const size_t base = (size_t)b * T_STEPS * S_ELEMS
                      + (size_t)sc * CHUNK + (size_t)tid * VEC;

    unsigned long long gaddr =
        (unsigned long long)(uintptr_t)(x + base);
    unsigned lds0 =
        (unsigned)(uintptr_t)(&smem[0][0]) + (unsigned)tid * (unsigned)sizeof(v4f);

    // Fire all 16 timestep tiles into LDS through the async data mover.
    // Per-lane 16B, per-wave 512B per stage; ASYNCcnt tracks completion in
    // issue order, so stage t is resident once ASYNCcnt <= 15 - t.
#pragma unroll
    for (int t = 0; t < T_STEPS; ++t) {
        asm volatile("global_load_async_to_lds_b128 %0, %1, off th:TH_LOAD_NT"
                     :: "v"(lds0 + (unsigned)t * (unsigned)sizeof(smem[0])),
                        "v"(gaddr + (unsigned long long)t *
                            ((unsigned long long)S_ELEMS * sizeof(float)))
                     : "memory");
    }

    v4f* gout = (v4f*)(out + base);
    v4f  v = (v4f)(0.0f);   // V_REST

#define LIF_STEP(t, rem)                                                      \
    {                                                                         \
        asm volatile("s_wait_asynccnt " #rem ::: "memory");                   \
        v4f xt = smem[t][tid];                                                \
        v.x = fmaf(v.x, 0.9f, xt.x);                                          \
        v.y = fmaf(v.y, 0.9f, xt.y);                                          \
        v.z = fmaf(v.z, 0.9f, xt.z);                                          \
        v.w = fmaf(v.w, 0.9f, xt.w);                                          \
        v4f sp;                                                               \
        sp.x = (v.x >= 1.0f) ? 1.0f : 0.0f;                                   \
        sp.y = (v.y >= 1.0f) ? 1.0f : 0.0f;                                   \
        sp.z = (v.z >= 1.0f) ? 1.0f : 0.0f;                                   \
        sp.w = (v.w >= 1.0f) ? 1.0f : 0.0f;                                   \
        v.x -= sp.x; v.y -= sp.y; v.z -= sp.z; v.w -= sp.w;                   \
        __builtin_nontemporal_store(sp, gout + (size_t)(t) * (S_ELEMS / VEC));\
    }

    LIF_STEP(0, 15)  LIF_STEP(1, 14)  LIF_STEP(2, 13)  LIF_STEP(3, 12)
    LIF_STEP(4, 11)  LIF_STEP(5, 10)  LIF_STEP(6, 9)   LIF_STEP(7, 8)
    LIF_STEP(8, 7)   LIF_STEP(9, 6)   LIF_STEP(10, 5)  LIF_STEP(11, 4)
    LIF_STEP(12, 3)  LIF_STEP(13, 2)  LIF_STEP(14, 1)  LIF_STEP(15, 0)
#undef LIF_STEP
}

extern "C" void kernel_launch(void* const* d_in, const int* in_sizes, int n_in,
                              void* d_out, int out_size, void* d_ws, size_t ws_size,
                              hipStream_t stream) {
    (void)in_sizes; (void)n_in; (void)out_size; (void)d_ws; (void)ws_size;
    const float* x   = (const float*)d_in[0];
    float*       out = (float*)d_out;

    const int nblocks = (B_BATCH * S_ELEMS) / CHUNK;  // 2048
    lif_scan_async<<<nblocks, NTHREADS, 0, stream>>>(x, out);
}